// ParameterHead_68281390071814
// MI455X (gfx1250) — compile-verified
//
#include <hip/hip_runtime.h>
#include <hip/hip_bf16.h>
#include <math.h>

typedef __attribute__((ext_vector_type(2))) float v2f;
typedef __attribute__((ext_vector_type(8))) float v8f;

#define ALPHA  0.9f
#define LN_EPS 1e-5f

// ---------------------------------------------------------------------------
// Kernel A: temporal pooling  P[b,n,d] = sum_t w_t * nf[b,t,n,d]
// Pure HBM stream: 262 MB read, 4 MB write -> ~11 us roofline on MI455X.
// ---------------------------------------------------------------------------
__global__ void __launch_bounds__(256)
pool_time_kernel(const float* __restrict__ nf, float* __restrict__ pooled,
                 int B, int T, int N, int D) {
  const int d4 = D >> 2;                               // float4 per row
  long long totalV4 = (long long)B * N * d4;
  long long idx = (long long)blockIdx.x * blockDim.x + threadIdx.x;
  if (idx >= totalV4) return;
  long long bn = idx / d4;
  int dv = (int)(idx % d4);
  long long b = bn / N;
  long long n = bn % N;
  const long long strideT = (long long)N * d4;         // float4 stride between t
  const float4* base = (const float4*)nf + b * (long long)T * strideT + n * d4 + dv;

  // w_t = alpha^(T-1-t) / norm ; iterate t = T-1 .. 0 so w starts at 1/norm
  float norm = (1.0f - powf(ALPHA, (float)T)) / (1.0f - ALPHA);
  float w = 1.0f / norm;
  float4 acc = make_float4(0.f, 0.f, 0.f, 0.f);
  for (int t = T - 1; t >= 0; --t) {
    const float4* p = base + (long long)t * strideT;
    if (t > 0) __builtin_prefetch((const void*)(p - strideT), 0, 1);
    float4 v = *p;
    acc.x += w * v.x; acc.y += w * v.y; acc.z += w * v.z; acc.w += w * v.w;
    w *= ALPHA;
  }
  ((float4*)pooled)[bn * d4 + dv] = acc;
}

// ---------------------------------------------------------------------------
// Wave-level 16xNN = 16xK @ KxNN GEMM using V_WMMA_F32_16X16X4_F32.
// A is LDS-resident row-major [16][K]; W is global row-major [K][NN].
// Output (bias, optional ReLU applied) stored to LDS row-major [16][NN].
// Fragment layouts per CDNA5 ISA 7.12.2 (32-bit A 16x4 / 32-bit C 16x16).
// ---------------------------------------------------------------------------
template <int K, int NN, int NT, bool RELU>
__device__ __forceinline__ void
wave_gemm16(const float* __restrict__ Alds, const float* __restrict__ W,
            const float* __restrict__ bias, float* __restrict__ Xout, int lane) {
  const int half = lane >> 4;       // 0: lanes 0-15, 1: lanes 16-31
  const int m = lane & 15;
  v8f acc[NT] = {};
#pragma unroll 1
  for (int k = 0; k < K / 4; ++k) {
    // A frag: VGPR0 = K(4k + 2*half), VGPR1 = K(4k + 2*half + 1), row M = m
    v2f a;
    const float* ap = Alds + m * K + 4 * k + 2 * half;
    a.x = ap[0];
    a.y = ap[1];
#pragma unroll
    for (int n = 0; n < NT; ++n) {
      // B frag: same K split across halves, N = n*16 + m across lanes
      const int col = n * 16 + m;
      const int kr = 4 * k + 2 * half;
      v2f bf;
      bf.x = W[(long long)kr * NN + col];
      bf.y = W[(long long)(kr + 1) * NN + col];
      acc[n] = __builtin_amdgcn_wmma_f32_16x16x4_f32(
          /*neg_a=*/false, a, /*neg_b=*/false, bf,
          /*c_mod=*/(short)0, acc[n], /*reuse_a=*/false, /*reuse_b=*/false);
    }
  }
  // C layout: VGPR r -> M = r + 8*half, N = n*16 + m
#pragma unroll
  for (int n = 0; n < NT; ++n) {
    float bv = bias[n * 16 + m];
#pragma unroll
    for (int r = 0; r < 8; ++r) {
      float v = acc[n][r] + bv;
      if (RELU) v = fmaxf(v, 0.0f);
      Xout[(r + 8 * half) * NN + n * 16 + m] = v;
    }
  }
}

template <int L>
__device__ __forceinline__ void
wave_layernorm16(float* __restrict__ X, const float* __restrict__ g,
                 const float* __restrict__ be, int lane) {
  if (lane < 16) {
    float* row = X + lane * L;
    float mu = 0.f;
    for (int c = 0; c < L; ++c) mu += row[c];
    mu *= (1.0f / L);
    float var = 0.f;
    for (int c = 0; c < L; ++c) { float d = row[c] - mu; var += d * d; }
    var *= (1.0f / L);
    float inv = rsqrtf(var + LN_EPS);
    for (int c = 0; c < L; ++c) row[c] = (row[c] - mu) * inv * g[c] + be[c];
  }
}

__device__ __forceinline__ float softplusf(float x) {
  return (x > 20.0f) ? x : log1pf(expf(x));
}

// ---------------------------------------------------------------------------
// Kernel B: per 16-edge tile (one wave32 per workgroup):
//   gather P[src]||P[dst] (16x256) -> @W_edge+b -> @W1+b1,ReLU,LN
//   -> @W2+b2,ReLU,LN -> @W3+b3 -> softplus -> out[2][B][E]
// ---------------------------------------------------------------------------
__global__ void __launch_bounds__(32)
mlp_head_kernel(const float* __restrict__ pooled,       // [B][N][128]
                const long long* __restrict__ edge_index, // [2][E] int64
                const float* __restrict__ W_edge, const float* __restrict__ b_edge,
                const float* __restrict__ W1, const float* __restrict__ b1,
                const float* __restrict__ g1, const float* __restrict__ be1,
                const float* __restrict__ W2, const float* __restrict__ b2,
                const float* __restrict__ g2, const float* __restrict__ be2,
                const float* __restrict__ W3, const float* __restrict__ b3,
                float* __restrict__ out, int B, int N, int E) {
  __shared__ float Abuf[16 * 256];   // 16 KB: GEMM1 input, reused as GEMM2 output
  __shared__ float Xbuf[16 * 128];   // 8 KB : GEMM1 output, reused as GEMM3 output
  const int lane = threadIdx.x;
  const int tilesPerB = E / 16;
  const int b = blockIdx.x / tilesPerB;
  const int e0 = (blockIdx.x % tilesPerB) * 16;
  const float* poolB = pooled + (long long)b * N * 128;

  // Stage A: 16 rows x 256 f32 = 1024 float4, fully coalesced gather from L2.
  for (int f = lane; f < 1024; f += 32) {
    int row = f >> 6;
    int c4 = (f & 63) * 4;
    long long node = (c4 < 128) ? edge_index[e0 + row] : edge_index[E + e0 + row];
    int col = (c4 < 128) ? c4 : (c4 - 128);
    float4 v = *(const float4*)(poolB + node * 128 + col);
    *(float4*)&Abuf[row * 256 + c4] = v;
  }
  __syncthreads();

  // edge_feat (already temporally pooled): 16x256 @ 256x128 + b_edge
  wave_gemm16<256, 128, 8, false>(Abuf, W_edge, b_edge, Xbuf, lane);
  __syncthreads();
  // h1 = LN(relu(X @ W1 + b1)) : 16x128 @ 128x128
  wave_gemm16<128, 128, 8, true>(Xbuf, W1, b1, Abuf, lane);
  __syncthreads();
  wave_layernorm16<128>(Abuf, g1, be1, lane);
  __syncthreads();
  // h2 = LN(relu(h1 @ W2 + b2)) : 16x128 @ 128x64
  wave_gemm16<128, 64, 4, true>(Abuf, W2, b2, Xbuf, lane);
  __syncthreads();
  wave_layernorm16<64>(Xbuf, g2, be2, lane);
  __syncthreads();
  // out = softplus(h2 @ W3 + b3), W3 is 64x2 -> scalar matvec per row
  if (lane < 16) {
    const float* row = Xbuf + lane * 64;
    float o0 = b3[0], o1 = b3[1];
    for (int c = 0; c < 64; ++c) {
      float h = row[c];
      o0 += h * W3[2 * c + 0];
      o1 += h * W3[2 * c + 1];
    }
    long long e = e0 + lane;
    out[(long long)b * E + e] = softplusf(o0);                       // r_line
    out[(long long)B * E + (long long)b * E + e] = softplusf(o1);    // x_line
  }
}

extern "C" void kernel_launch(void* const* d_in, const int* in_sizes, int n_in,
                              void* d_out, int out_size, void* d_ws, size_t ws_size,
                              hipStream_t stream) {
  const float*     nf     = (const float*)d_in[0];
  const long long* ei     = (const long long*)d_in[1];   // int64 edge_index
  const float*     W_edge = (const float*)d_in[2];
  const float*     b_edge = (const float*)d_in[3];
  const float*     W1     = (const float*)d_in[4];
  const float*     b1     = (const float*)d_in[5];
  const float*     g1     = (const float*)d_in[6];
  const float*     be1    = (const float*)d_in[7];
  const float*     W2     = (const float*)d_in[8];
  const float*     b2     = (const float*)d_in[9];
  const float*     g2     = (const float*)d_in[10];
  const float*     be2    = (const float*)d_in[11];
  const float*     W3     = (const float*)d_in[12];
  const float*     b3     = (const float*)d_in[13];
  float*           outp   = (float*)d_out;

  const int D = 128;
  const int T = 64;
  const int E = in_sizes[1] / 2;                 // 4096
  const int B = out_size / (2 * E);              // 4
  const int N = (int)((long long)in_sizes[0] / ((long long)B * T * D)); // 2000

  float* pooled = (float*)d_ws;                  // B*N*D floats = 4 MB

  long long totalV4 = (long long)B * N * (D / 4);
  int blocksA = (int)((totalV4 + 255) / 256);
  pool_time_kernel<<<blocksA, 256, 0, stream>>>(nf, pooled, B, T, N, D);

  int blocksB = B * (E / 16);                    // 1024 tiles, one wave32 each
  mlp_head_kernel<<<blocksB, 32, 0, stream>>>(pooled, ei,
                                              W_edge, b_edge,
                                              W1, b1, g1, be1,
                                              W2, b2, g2, be2,
                                              W3, b3,
                                              outp, B, N, E);
}